// FourierBlock2D_87016037417116
// MI455X (gfx1250) — compile-verified
//
#include <hip/hip_runtime.h>
#include <math.h>

// Problem constants (match reference)
#define B_    4
#define CIN_  64
#define COUT_ 64
#define NX_   256
#define NY_   256
#define M1_   20
#define M2_   20
#define KSEL_ 40              // 2 * M1 retained kx bands
#define PI_   3.14159265358979323846f
#define IMG_  (NX_ * NY_)     // 65536

typedef __attribute__((ext_vector_type(2))) float v2f;
typedef __attribute__((ext_vector_type(8))) float v8f;

// ---------------------------------------------------------------------------
// Kernel 1: partial rfft along Y.  One block per (b, i, xrow).
// X1[b][i][x][ky] = sum_y x[...][y] * e^{-i 2pi ky y / 256},  ky in [0,20)
// ---------------------------------------------------------------------------
__global__ void k_ydft(const float* __restrict__ x, float* __restrict__ X1) {
    __shared__ float  row[NY_];
    __shared__ float2 tw[256];
    const int rowid = blockIdx.x;                     // b*CIN*NX + i*NX + xr
    const float* rp = x + (size_t)rowid * NY_;
    for (int t = threadIdx.x; t < 256; t += blockDim.x) {
        float ang = (2.0f * PI_ / 256.0f) * (float)t;
        tw[t] = make_float2(cosf(ang), sinf(ang));
        row[t] = rp[t];
    }
    __syncthreads();
    const int ky = threadIdx.x;
    if (ky < M2_) {
        float cr = 0.0f, ci = 0.0f;
        for (int y = 0; y < NY_; ++y) {
            float  v = row[y];
            float2 w = tw[(ky * y) & 255];
            cr += v * w.x;
            ci -= v * w.y;
        }
        ((float2*)X1)[(size_t)rowid * M2_ + ky] = make_float2(cr, ci);
    }
}

// ---------------------------------------------------------------------------
// Kernel 2: partial DFT along X for retained kx.  One block per (b, i, ky).
// Xft[b][i][sel][ky] = sum_x X1[b][i][x][ky] * e^{-i 2pi kx x / 256}
// sel 0..19 -> kx = sel ; sel 20..39 -> kx = 236 + (sel-20)
// ---------------------------------------------------------------------------
__global__ void k_xdft(const float* __restrict__ X1, float* __restrict__ Xft) {
    __shared__ float2 col[NX_];
    __shared__ float2 tw[256];
    const int id = blockIdx.x;                        // (b*CIN + i)*M2 + ky
    const int ky = id % M2_;
    const int bi = id / M2_;
    const float2* src = (const float2*)X1 + (size_t)bi * NX_ * M2_ + ky;
    for (int t = threadIdx.x; t < 256; t += blockDim.x) {
        float ang = (2.0f * PI_ / 256.0f) * (float)t;
        tw[t]  = make_float2(cosf(ang), sinf(ang));
        col[t] = src[(size_t)t * M2_];
    }
    __syncthreads();
    const int s = threadIdx.x;
    if (s < KSEL_) {
        const int kx = (s < M1_) ? s : (NX_ - M1_ + (s - M1_));
        float cr = 0.0f, ci = 0.0f;
        for (int xx = 0; xx < NX_; ++xx) {
            float2 v = col[xx];
            float2 w = tw[(kx * xx) & 255];
            cr += v.x * w.x + v.y * w.y;              // v * conj-twiddle
            ci += v.y * w.x - v.x * w.y;
        }
        ((float2*)Xft)[((size_t)bi * KSEL_ + s) * M2_ + ky] = make_float2(cr, ci);
    }
}

// ---------------------------------------------------------------------------
// Kernel 3: spectral channel mix.  One block per (b, mode); thread = o.
// Oft[b][o][sel][ky] = sum_i Xft[b][i][sel][ky] * (wr + i wi)[i][o][kx][ky]
// ---------------------------------------------------------------------------
__global__ void k_specmul(const float* __restrict__ Xft,
                          const float* __restrict__ w1r, const float* __restrict__ w1i,
                          const float* __restrict__ w2r, const float* __restrict__ w2i,
                          float* __restrict__ Oft) {
    const int id  = blockIdx.x;                       // b*800 + m
    const int m   = id % (KSEL_ * M2_);
    const int b   = id / (KSEL_ * M2_);
    const int sel = m / M2_;
    const int ky  = m % M2_;
    const int o   = threadIdx.x;                      // 0..63
    const float* wr;
    const float* wi;
    int kx;
    if (sel < M1_) { wr = w1r; wi = w1i; kx = sel; }
    else           { wr = w2r; wi = w2i; kx = sel - M1_; }
    const int wbase = kx * M2_ + ky;
    float cr = 0.0f, ci = 0.0f;
    for (int i = 0; i < CIN_; ++i) {
        float2 v  = ((const float2*)Xft)[((size_t)(b * CIN_ + i) * KSEL_ + sel) * M2_ + ky];
        float  wwr = wr[(size_t)(i * COUT_ + o) * (M1_ * M2_) + wbase];
        float  wwi = wi[(size_t)(i * COUT_ + o) * (M1_ * M2_) + wbase];
        cr += v.x * wwr - v.y * wwi;
        ci += v.x * wwi + v.y * wwr;
    }
    ((float2*)Oft)[((size_t)(b * COUT_ + o) * KSEL_ + sel) * M2_ + ky] = make_float2(cr, ci);
}

// ---------------------------------------------------------------------------
// Kernel 4: inverse DFT along X.  One block per (b, o, ky); thread = x.
// Y1[b][o][x][ky] = sum_sel Oft[b][o][sel][ky] * e^{+i 2pi kx x / 256}
// ---------------------------------------------------------------------------
__global__ void k_invx(const float* __restrict__ Oft, float* __restrict__ Y1) {
    __shared__ float2 modes[KSEL_];
    __shared__ float2 tw[256];
    const int id = blockIdx.x;                        // (b*COUT + o)*M2 + ky
    const int ky = id % M2_;
    const int bo = id / M2_;
    for (int t = threadIdx.x; t < 256; t += blockDim.x) {
        float ang = (2.0f * PI_ / 256.0f) * (float)t;
        tw[t] = make_float2(cosf(ang), sinf(ang));
    }
    if (threadIdx.x < KSEL_)
        modes[threadIdx.x] =
            ((const float2*)Oft)[((size_t)bo * KSEL_ + threadIdx.x) * M2_ + ky];
    __syncthreads();
    const int xx = threadIdx.x;                       // 0..255
    float cr = 0.0f, ci = 0.0f;
    for (int s = 0; s < KSEL_; ++s) {
        const int kx = (s < M1_) ? s : (NX_ - M1_ + (s - M1_));
        float2 v = modes[s];
        float2 w = tw[(kx * xx) & 255];
        cr += v.x * w.x - v.y * w.y;
        ci += v.x * w.y + v.y * w.x;
    }
    ((float2*)Y1)[((size_t)bo * NX_ + xx) * M2_ + ky] = make_float2(cr, ci);
}

// ---------------------------------------------------------------------------
// Kernel 5 (fused, WMMA): per wave one 16(o) x 16(y) output tile at (b, xrow).
//   linear part: C = lin_w[otile, :] (16x64) @ x[b, :, xrow, ytile:+16] (64x16)
//   via 16x V_WMMA_F32_16X16X4_F32.
//   spectral part: irfft along Y from Y1 modes, + bias, tanh, store.
// ---------------------------------------------------------------------------
__global__ void k_fuse(const float* __restrict__ x,  const float* __restrict__ Y1,
                       const float* __restrict__ lw, const float* __restrict__ lb,
                       float* __restrict__ out) {
    __shared__ float2 tw[256];
    for (int t = threadIdx.x; t < 256; t += blockDim.x) {
        float ang = (2.0f * PI_ / 256.0f) * (float)t;
        tw[t] = make_float2(cosf(ang), sinf(ang));
    }
    __syncthreads();

    const int wid   = blockIdx.x * (blockDim.x >> 5) + (threadIdx.x >> 5);
    const int lane  = threadIdx.x & 31;
    const int t     = wid & 63;                       // tile within (b,xrow)
    const int rowid = wid >> 6;                       // b*NX + xr
    const int b     = rowid >> 8;
    const int xr    = rowid & 255;
    const int otile = (t >> 4) * 16;
    const int ytile = (t & 15) * 16;

    const int mn    = lane & 15;                      // A: row M ; B: col N
    const int khalf = (lane >> 4) * 2;                // K split across lane halves

    v8f c = {};
    const float* xb = x + (size_t)b * CIN_ * IMG_ + (size_t)xr * NY_ + ytile;
    for (int k0 = 0; k0 < CIN_; k0 += 4) {
        v2f a, bb;
        // A (16x4 f32): lanes 0-15 M=lane, K = k0+{0,1}; lanes 16-31 K = k0+{2,3}
        const int arow = otile + mn;
        a.x = lw[arow * CIN_ + k0 + khalf];
        a.y = lw[arow * CIN_ + k0 + khalf + 1];
        // B (4x16 f32): VGPR0 rows K = k0+{0|2}, VGPR1 rows K = k0+{1|3}
        const float* xp = xb + (size_t)(k0 + khalf) * IMG_ + mn;
        bb.x = xp[0];
        bb.y = xp[IMG_];
        c = __builtin_amdgcn_wmma_f32_16x16x4_f32(false, a, false, bb,
                                                  (short)0, c, false, false);
    }

    // Finalize: C layout = VGPR r holds rows M=r (lanes 0-15) / M=r+8 (lanes 16-31),
    // column N = lane&15.
    const int   y  = ytile + mn;
    const int   hi = lane >> 4;
    const float sc = 1.0f / (float)IMG_;              // irfft2 1/(NX*NY)
    for (int r = 0; r < 8; ++r) {
        const int o = otile + r + 8 * hi;
        const float2* mp =
            (const float2*)Y1 + ((size_t)(b * COUT_ + o) * NX_ + xr) * M2_;
        float acc = mp[0].x;                          // Re(C0)
        for (int ky = 1; ky < M2_; ++ky) {
            float2 v = mp[ky];
            float2 w = tw[(ky * y) & 255];
            acc += 2.0f * (v.x * w.x - v.y * w.y);    // 2*Re(Ck * e^{+i th})
        }
        const float val = acc * sc + c[r] + lb[o];
        out[((size_t)(b * COUT_ + o) * NX_ + xr) * NY_ + y] = tanhf(val);
    }
}

// ---------------------------------------------------------------------------
extern "C" void kernel_launch(void* const* d_in, const int* in_sizes, int n_in,
                              void* d_out, int out_size, void* d_ws, size_t ws_size,
                              hipStream_t stream) {
    const float* x   = (const float*)d_in[0];
    const float* w1r = (const float*)d_in[1];
    const float* w1i = (const float*)d_in[2];
    const float* w2r = (const float*)d_in[3];
    const float* w2i = (const float*)d_in[4];
    const float* lw  = (const float*)d_in[5];
    const float* lb  = (const float*)d_in[6];
    float* out = (float*)d_out;

    // Workspace carve-up (floats). Total ~24.3 MB, L2-resident.
    float* X1  = (float*)d_ws;                                   // B*CIN*NX*M2*2
    float* Xft = X1  + (size_t)B_ * CIN_ * NX_ * M2_ * 2;        // B*CIN*40*M2*2
    float* Oft = Xft + (size_t)B_ * CIN_ * KSEL_ * M2_ * 2;      // B*COUT*40*M2*2
    float* Y1  = Oft + (size_t)B_ * COUT_ * KSEL_ * M2_ * 2;     // B*COUT*NX*M2*2

    k_ydft   <<<B_ * CIN_ * NX_,       64, 0, stream>>>(x, X1);
    k_xdft   <<<B_ * CIN_ * M2_,       64, 0, stream>>>(X1, Xft);
    k_specmul<<<B_ * KSEL_ * M2_,      64, 0, stream>>>(Xft, w1r, w1i, w2r, w2i, Oft);
    k_invx   <<<B_ * COUT_ * M2_,     256, 0, stream>>>(Oft, Y1);
    // 65536 wave-tiles, 8 waves (256 threads) per block
    k_fuse   <<<(B_ * NX_ * 64) / 8,  256, 0, stream>>>(x, Y1, lw, lb, out);
}